// ItemEmbeddingModule_35828617183451
// MI455X (gfx1250) — compile-verified
//
#include <hip/hip_runtime.h>
#include <hip/hip_bf16.h>

// Problem constants (match reference): B=4096, F=32, V=100000, L=20, E=64
#define BB 4096
#define FF 32
#define VV 100000
#define LL 20
#define EE 64

#define WAVES_PER_BLOCK 8
#define ROW_BYTES (EE * 4)          // 256 B per embedding row

// One wave32 per bag (b, f).
// 1) One lane-parallel load grabs all 20 bag ids (lane l holds ids[l]).
// 2) Async gather: GLOBAL_LOAD_ASYNC_TO_LDS_B128 moves 16B/lane -> 512B
//    = 2 embedding rows per instruction; ids delivered to lanes via __shfl
//    (ds_bpermute), so the issue loop never touches VMEM latency.
// 3) s_wait_asynccnt 0, then sum-reduce rows out of LDS (float2 per lane).
__global__ __launch_bounds__(256) void emb_bag_sum_kernel(
    const int*   __restrict__ ids,      // [B, F, L]
    const int*   __restrict__ lengths,  // [B, F]
    const float* __restrict__ tables,   // [F, V, E]
    float*       __restrict__ out)      // [B, F*E]
{
    // 8 waves * 20 rows * 256B = 40 KB
    __shared__ float sh[WAVES_PER_BLOCK * LL * EE];

    const int lane = threadIdx.x & 31;
    const int wave = threadIdx.x >> 5;

    // bag = f*B + b  -> 512 consecutive blocks share the same table f
    const int bag = blockIdx.x * WAVES_PER_BLOCK + wave;
    const int f   = bag >> 12;          // bag / 4096
    const int b   = bag & (BB - 1);     // bag % 4096

    const int* __restrict__ bagIds = ids + ((size_t)b * FF + f) * (size_t)LL;
    int len = lengths[(size_t)b * FF + f];
    len = len < 0 ? 0 : (len > LL ? LL : len);

    // Preload all bag ids: lane l (< 20) owns ids[l].
    int myId = 0;
    if (lane < LL) myId = bagIds[lane];

    // --- async gather: 2 rows per instruction, 16B per lane ---
    const int sub  = lane & 15;         // 16-byte chunk index within a row
    const int half = lane >> 4;         // 0 -> row 2j, 1 -> row 2j+1

    // Low 32 bits of a flat LDS address == in-LDS byte offset (VDST operand).
    char* ldsSlab = (char*)sh + (size_t)wave * (LL * ROW_BYTES);
    const uint64_t laneGlobalBase =
        (uint64_t)(uintptr_t)tables
        + (uint64_t)f * (uint64_t)VV * ROW_BYTES
        + (uint64_t)(sub * 16);
    uint32_t la = (uint32_t)(uintptr_t)(ldsSlab + half * ROW_BYTES + sub * 16);

    const int pairs = (len + 1) >> 1;   // extra odd row is a valid, unused gather
    for (int j = 0; j < pairs; ++j) {
        const int srcLane = 2 * j + half;            // <= 19 always
        const int id = __shfl(myId, srcLane);        // ds_bpermute_b32
        const uint64_t ga = laneGlobalBase + (uint64_t)id * ROW_BYTES;
        asm volatile("global_load_async_to_lds_b128 %0, %1, off"
                     :: "v"(la), "v"(ga)
                     : "memory");
        la += 2 * ROW_BYTES;                         // next row pair
    }

#if __has_builtin(__builtin_amdgcn_s_wait_asynccnt)
    __builtin_amdgcn_s_wait_asynccnt(0);
#else
    asm volatile("s_wait_asynccnt 0" ::: "memory");
#endif

    // --- reduce from LDS: lane owns columns (2*lane, 2*lane+1) of E=64 ---
    const float2* __restrict__ shv =
        (const float2*)(sh + wave * (LL * EE)) + lane;   // ds_load_b64, no conflicts
    float accx = 0.0f, accy = 0.0f;
    #pragma unroll 5
    for (int l = 0; l < len; ++l) {
        const float2 v = shv[l * (EE / 2)];
        accx += v.x;
        accy += v.y;
    }

    // nan_to_num (reference applies nan-fill)
    if (isnan(accx)) accx = 0.0f;
    if (isnan(accy)) accy = 0.0f;

    float2 acc = make_float2(accx, accy);
    float2* o = (float2*)(out + ((size_t)b * FF + f) * (size_t)EE) + lane;
    *o = acc;
}

extern "C" void kernel_launch(void* const* d_in, const int* in_sizes, int n_in,
                              void* d_out, int out_size, void* d_ws, size_t ws_size,
                              hipStream_t stream) {
    (void)in_sizes; (void)n_in; (void)out_size; (void)d_ws; (void)ws_size;

    const int*   ids     = (const int*)d_in[0];    // [B,F,L]
    const int*   lengths = (const int*)d_in[1];    // [B,F]
    const float* tables  = (const float*)d_in[2];  // [F,V,E]
    float*       out     = (float*)d_out;          // [B, F*E]

    const int totalBags = BB * FF;                       // 131072 waves
    const int blocks    = totalBags / WAVES_PER_BLOCK;   // 16384 blocks of 256
    emb_bag_sum_kernel<<<blocks, WAVES_PER_BLOCK * 32, 0, stream>>>(
        ids, lengths, tables, out);
}